// GNN_9680856285887
// MI455X (gfx1250) — compile-verified
//
#include <hip/hip_runtime.h>
#include <hip/hip_bf16.h>

typedef float v2f __attribute__((ext_vector_type(2)));
typedef float v8f __attribute__((ext_vector_type(8)));

#define D_FEAT 128

// ---------------------------------------------------------------------------
// GEMM: H[M x 128] = A[M x 128] * W[128 x 128] via V_WMMA_F32_16X16X4_F32.
// Block = 256 threads = 8 waves; block owns a 16-row stripe staged once in
// LDS (8 KB); wave w owns the 16x16 output tile at columns [16w,16w+16).
// A fragment layout (16x4 f32): lanes 0-15 -> M=0..15; lane>=16 selects K+2/3.
// C/D (16x16 f32): vgpr i holds M=i (lanes 0-15) / M=i+8 (lanes 16-31).
// ---------------------------------------------------------------------------
__global__ void k_gemm16(const float* __restrict__ A, const float* __restrict__ W,
                         float* __restrict__ H, int nrows) {
  __shared__ float Atile[16 * D_FEAT];        // 8 KB
  const int row0 = blockIdx.x * 16;
  const int tid  = threadIdx.x;

  // Cooperative stage of the 16x128 A stripe (rows clamped for tail block).
  for (int t = tid; t < 16 * D_FEAT; t += 256) {
    int r = row0 + (t >> 7);
    if (r > nrows - 1) r = nrows - 1;
    Atile[t] = A[(size_t)r * D_FEAT + (t & (D_FEAT - 1))];
  }
  __syncthreads();

  const int wave = tid >> 5;
  const int lane = tid & 31;
  const int col0 = wave << 4;
  const int l    = lane & 15;
  const int hi   = lane >> 4;                 // 0: K+0/1, 1: K+2/3
  const float* Ash = Atile + l * D_FEAT;
  const float* Wp  = W + col0 + l;

  v8f c = {};
#pragma unroll 8
  for (int k = 0; k < D_FEAT; k += 4) {
    const int ka = k + (hi << 1);
    v2f a, b;
    a.x = Ash[ka];                            // ds_load (LDS)
    a.y = Ash[ka + 1];
    b.x = Wp[(size_t)ka * D_FEAT];
    b.y = Wp[(size_t)(ka + 1) * D_FEAT];
    c = __builtin_amdgcn_wmma_f32_16x16x4_f32(false, a, false, b,
                                              (short)0, c, false, false);
  }

  const int mbase = hi << 3;
  const int col   = col0 + l;
  float* Hp = H + (size_t)(row0 + mbase) * D_FEAT + col;
  if (row0 + 16 <= nrows) {                   // fast path: full stripe
#pragma unroll
    for (int i = 0; i < 8; ++i) Hp[(size_t)i * D_FEAT] = c[i];
  } else {
#pragma unroll
    for (int i = 0; i < 8; ++i)
      if (row0 + mbase + i < nrows) Hp[(size_t)i * D_FEAT] = c[i];
  }
}

// ---------------------------------------------------------------------------
// Per-node attention scores: a_s[n] = h[n].att_src, a_d[n] = h[n].att_dst.
// ---------------------------------------------------------------------------
__global__ void k_scores(const float* __restrict__ H, const float* __restrict__ as,
                         const float* __restrict__ ad, float* __restrict__ outs,
                         float* __restrict__ outd, int N) {
  const int nid  = (int)((blockIdx.x * (unsigned)blockDim.x + threadIdx.x) >> 5);
  const int lane = threadIdx.x & 31;
  if (nid >= N) return;
  const float* hrow = H + (size_t)nid * D_FEAT;
  float ss = 0.f, dd = 0.f;
#pragma unroll
  for (int j = 0; j < 4; ++j) {
    const int idx = lane + 32 * j;
    const float hv = hrow[idx];
    ss += hv * as[idx];
    dd += hv * ad[idx];
  }
#pragma unroll
  for (int off = 16; off > 0; off >>= 1) {
    ss += __shfl_down(ss, off, 32);
    dd += __shfl_down(dd, off, 32);
  }
  if (lane == 0) { outs[nid] = ss; outd[nid] = dd; }
}

// ---------------------------------------------------------------------------
// CSR-by-dst build (layer-invariant; built once).
// ---------------------------------------------------------------------------
__device__ __forceinline__ void load_edge(const int* __restrict__ ei, int E,
                                          int i, int& s, int& d) {
  if (i < E) { s = ei[i]; d = ei[E + i]; }
  else       { s = i - E; d = i - E; }        // self loops appended
}

__global__ void k_fill_i(int* __restrict__ p, int n, int v) {
  const int i = blockIdx.x * blockDim.x + threadIdx.x;
  if (i < n) p[i] = v;
}

__global__ void k_count(const int* __restrict__ ei, int E, int ET,
                        int* __restrict__ deg) {
  const int i = blockIdx.x * blockDim.x + threadIdx.x;
  if (i >= ET) return;
  int s, d; load_edge(ei, E, i, s, d);
  atomicAdd(deg + d, 1);
}

// Per-block inclusive scan (LDS) -> exclusive rowstart + per-block totals.
__global__ void k_scan1(const int* __restrict__ deg, int* __restrict__ rowstart,
                        int* __restrict__ bsum, int N) {
  __shared__ int sh[256];
  const int tid = threadIdx.x;
  const int i   = blockIdx.x * 256 + tid;
  const int v   = (i < N) ? deg[i] : 0;
  sh[tid] = v;
  __syncthreads();
#pragma unroll
  for (int off = 1; off < 256; off <<= 1) {
    const int t = (tid >= off) ? sh[tid - off] : 0;
    __syncthreads();
    sh[tid] += t;
    __syncthreads();
  }
  if (i < N) rowstart[i] = sh[tid] - v;       // exclusive
  if (tid == 255) bsum[blockIdx.x] = sh[255];
}

// Single-block exclusive scan of block sums (handles arbitrary NB in chunks).
__global__ void k_scan2(int* __restrict__ bsum, int NB) {
  __shared__ int sh[256];
  __shared__ int roff;
  const int tid = threadIdx.x;
  if (tid == 0) roff = 0;
  __syncthreads();
  for (int base = 0; base < NB; base += 256) {
    const int i = base + tid;
    const int v = (i < NB) ? bsum[i] : 0;
    sh[tid] = v;
    __syncthreads();
#pragma unroll
    for (int off = 1; off < 256; off <<= 1) {
      const int t = (tid >= off) ? sh[tid - off] : 0;
      __syncthreads();
      sh[tid] += t;
      __syncthreads();
    }
    if (i < NB) bsum[i] = (sh[tid] - v) + roff;
    __syncthreads();
    if (tid == 0) roff += sh[255];
    __syncthreads();
  }
}

__global__ void k_scan3(int* __restrict__ rowstart, const int* __restrict__ bsum,
                        int* __restrict__ cursor, int N, int ET) {
  const int i = blockIdx.x * blockDim.x + threadIdx.x;
  if (i >= N) return;
  const int r = rowstart[i] + bsum[i >> 8];
  rowstart[i] = r;
  cursor[i]   = r;
  if (i == 0) rowstart[N] = ET;
}

__global__ void k_scatter(const int* __restrict__ ei, int E, int ET,
                          int* __restrict__ cursor, int* __restrict__ csr_src) {
  const int i = blockIdx.x * blockDim.x + threadIdx.x;
  if (i >= ET) return;
  int s, d; load_edge(ei, E, i, s, d);
  const int pos = atomicAdd(cursor + d, 1);
  csr_src[pos] = s;
}

// ---------------------------------------------------------------------------
// Fused per-dst softmax + aggregation + bias (+optional leaky_relu 0.01).
// Wave per dst node; three sweeps over incident edges; NO float atomics.
// ---------------------------------------------------------------------------
__device__ __forceinline__ float leaky02(float v) {
  return v > 0.f ? v : 0.2f * v;
}

__global__ void k_node_agg(const int* __restrict__ rowstart,
                           const int* __restrict__ csr_src,
                           const float* __restrict__ a_s,
                           const float* __restrict__ a_d,
                           const float* __restrict__ H,
                           const float* __restrict__ bias,
                           float* __restrict__ O, int N, int act) {
  const int nid  = (int)((blockIdx.x * (unsigned)blockDim.x + threadIdx.x) >> 5);
  const int lane = threadIdx.x & 31;
  if (nid >= N) return;

  const int   beg  = rowstart[nid];
  const int   end  = rowstart[nid + 1];
  const float ad_v = a_d[nid];

  // sweep 1: segment max (lanes stride edges, butterfly reduce)
  float m = -3.0e38f;
  for (int j = beg + lane; j < end; j += 32)
    m = fmaxf(m, leaky02(a_s[csr_src[j]] + ad_v));
#pragma unroll
  for (int off = 16; off > 0; off >>= 1)
    m = fmaxf(m, __shfl_xor(m, off, 32));

  // sweep 2: sum of exp
  float den = 0.f;
  for (int j = beg + lane; j < end; j += 32)
    den += __expf(leaky02(a_s[csr_src[j]] + ad_v) - m);
#pragma unroll
  for (int off = 16; off > 0; off >>= 1)
    den += __shfl_xor(den, off, 32);
  const float inv_den = 1.0f / (den + 1e-16f);

  // sweep 3: whole wave walks edges together; 4 features per lane.
  // Software-prefetch the next edge's h[src] row (global_prefetch_b8) so the
  // L2 round-trip overlaps the current edge's FMAs (hottest loop: ~14.5M
  // iterations x 512B gather, L2-resident).
  float acc0 = 0.f, acc1 = 0.f, acc2 = 0.f, acc3 = 0.f;
  int s_next = (beg < end) ? csr_src[beg] : 0;
  for (int j = beg; j < end; ++j) {
    const int s_cur = s_next;
    if (j + 1 < end) {
      s_next = csr_src[j + 1];
      __builtin_prefetch(H + (size_t)s_next * D_FEAT + lane, 0, 3);
    }
    const float w = __expf(leaky02(a_s[s_cur] + ad_v) - m) * inv_den;
    const float* hs = H + (size_t)s_cur * D_FEAT;          // coalesced 512B
    acc0 += w * hs[lane];
    acc1 += w * hs[lane + 32];
    acc2 += w * hs[lane + 64];
    acc3 += w * hs[lane + 96];
  }

  float* od = O + (size_t)nid * D_FEAT;
  float r0 = acc0 + bias[lane];
  float r1 = acc1 + bias[lane + 32];
  float r2 = acc2 + bias[lane + 64];
  float r3 = acc3 + bias[lane + 96];
  if (act) {
    r0 = r0 > 0.f ? r0 : 0.01f * r0;
    r1 = r1 > 0.f ? r1 : 0.01f * r1;
    r2 = r2 > 0.f ? r2 : 0.01f * r2;
    r3 = r3 > 0.f ? r3 : 0.01f * r3;
  }
  od[lane]      = r0;
  od[lane + 32] = r1;
  od[lane + 64] = r2;
  od[lane + 96] = r3;
}

// ---------------------------------------------------------------------------
// Global mean pool.
// ---------------------------------------------------------------------------
__global__ void k_fill(float* __restrict__ p, int n, float v) {
  const int i = blockIdx.x * blockDim.x + threadIdx.x;
  if (i < n) p[i] = v;
}

__global__ void k_pool(const float* __restrict__ O, const int* __restrict__ batch,
                       float* __restrict__ out, float* __restrict__ counts, int N) {
  const int nid  = (int)((blockIdx.x * (unsigned)blockDim.x + threadIdx.x) >> 5);
  const int lane = threadIdx.x & 31;
  if (nid >= N) return;
  const int g = batch[nid];
  const float* r = O + (size_t)nid * D_FEAT;
  float* og = out + (size_t)g * D_FEAT;
#pragma unroll
  for (int j = 0; j < 4; ++j) {
    const int idx = lane + 32 * j;
    atomicAdd(og + idx, r[idx]);          // global_atomic_add_f32
  }
  if (lane == 0) atomicAdd(counts + g, 1.0f);
}

__global__ void k_pool_div(float* __restrict__ out, const float* __restrict__ counts,
                           int n_elems) {
  const int i = blockIdx.x * blockDim.x + threadIdx.x;
  if (i >= n_elems) return;
  out[i] = out[i] / fmaxf(counts[i >> 7], 1.0f);
}

// ---------------------------------------------------------------------------
extern "C" void kernel_launch(void* const* d_in, const int* in_sizes, int n_in,
                              void* d_out, int out_size, void* d_ws, size_t ws_size,
                              hipStream_t stream) {
  const float* x     = (const float*)d_in[0];
  const int*   ei    = (const int*)d_in[1];
  const int*   batch = (const int*)d_in[2];
  const float* W1    = (const float*)d_in[3];
  const float* as1   = (const float*)d_in[4];
  const float* ad1   = (const float*)d_in[5];
  const float* b1    = (const float*)d_in[6];
  const float* W2    = (const float*)d_in[7];
  const float* as2   = (const float*)d_in[8];
  const float* ad2   = (const float*)d_in[9];
  const float* b2    = (const float*)d_in[10];
  float* out = (float*)d_out;

  const int N  = in_sizes[0] / D_FEAT;   // 50000
  const int E  = in_sizes[1] / 2;        // 800000
  const int ET = E + N;                  // + self loops
  const int NE = N * D_FEAT;
  const int NB = (N + 255) / 256;        // scan blocks

  char* wsp = (char*)d_ws;
  auto carve = [&](size_t bytes) {
    char* p = wsp;
    wsp += (bytes + 255) & ~(size_t)255;
    return p;
  };
  float* H        = (float*)carve((size_t)NE * 4);
  float* O        = (float*)carve((size_t)NE * 4);
  float* a_s      = (float*)carve((size_t)N * 4);
  float* a_d      = (float*)carve((size_t)N * 4);
  int*   deg      = (int*)carve((size_t)N * 4);
  int*   rowstart = (int*)carve((size_t)(N + 1) * 4);
  int*   cursor   = (int*)carve((size_t)N * 4);
  int*   bsum     = (int*)carve((size_t)NB * 4);
  int*   csr_src  = (int*)carve((size_t)ET * 4);
  float* cnts     = (float*)carve(64 * 4);

  const int TB = 256;

  // --- CSR-by-dst build (layer invariant) ---
  k_fill_i <<<(N + TB - 1) / TB, TB, 0, stream>>>(deg, N, 0);
  k_count  <<<(ET + TB - 1) / TB, TB, 0, stream>>>(ei, E, ET, deg);
  k_scan1  <<<NB, 256, 0, stream>>>(deg, rowstart, bsum, N);
  k_scan2  <<<1, 256, 0, stream>>>(bsum, NB);
  k_scan3  <<<(N + TB - 1) / TB, TB, 0, stream>>>(rowstart, bsum, cursor, N, ET);
  k_scatter<<<(ET + TB - 1) / TB, TB, 0, stream>>>(ei, E, ET, cursor, csr_src);

  auto layer = [&](const float* X, const float* W, const float* as,
                   const float* ad, const float* b, int act) {
    k_gemm16  <<<(N + 15) / 16, TB, 0, stream>>>(X, W, H, N);
    k_scores  <<<(N + 7) / 8, TB, 0, stream>>>(H, as, ad, a_s, a_d, N);
    k_node_agg<<<(N + 7) / 8, TB, 0, stream>>>(rowstart, csr_src, a_s, a_d,
                                               H, b, O, N, act);
  };

  layer(x, W1, as1, ad1, b1, /*act=*/1);   // layer 1 + leaky_relu(0.01)
  layer(O, W2, as2, ad2, b2, /*act=*/0);   // layer 2 (O -> H -> O)

  // --- global mean pool over 64 graphs ---
  k_fill    <<<(out_size + TB - 1) / TB, TB, 0, stream>>>(out, out_size, 0.0f);
  k_fill    <<<1, 64, 0, stream>>>(cnts, 64, 0.0f);
  k_pool    <<<(N + 7) / 8, TB, 0, stream>>>(O, batch, out, cnts, N);
  k_pool_div<<<(out_size + TB - 1) / TB, TB, 0, stream>>>(out, cnts, out_size);
}